// QuantumReservoir_65274912964827
// MI455X (gfx1250) — compile-verified
//
#include <hip/hip_runtime.h>

typedef float v2f __attribute__((ext_vector_type(2)));
typedef float v8f __attribute__((ext_vector_type(8)));

#define NQ    12
#define NL    4
#define DIM   4096      // 2^12 amplitudes
#define NPAIR 2048      // DIM/2 pairs per 1q gate
#define NWAVE 4         // waves per sample
#define NTILE 32        // WMMA tiles (16 pairs each) per wave per gate
#define PFD   4         // prefetch depth (must divide NTILE)

struct cpx { float r, i; };
__device__ __forceinline__ cpx cmul(cpx a, cpx b) { return { a.r*b.r - a.i*b.i, a.r*b.i + a.i*b.r }; }
__device__ __forceinline__ cpx cadd(cpx a, cpx b) { return { a.r + b.r, a.i + b.i }; }
__device__ __forceinline__ cpx cscl(cpx a, float s) { return { a.r*s, a.i*s }; }

__device__ __forceinline__ int pair_addr(int p, int bp, int lm, int off) {
    // expand pair id p by inserting a 0 bit at position bp, add s0/s1 offset
    return (((p >> bp) << (bp + 1)) | (p & lm)) + off;
}

// 4 waves (128 threads) per sample; 32KB state lives entirely in LDS.
__global__ __launch_bounds__(128) void qreservoir_kernel(const float* __restrict__ x,
                                                         const float* __restrict__ w,
                                                         float* __restrict__ out)
{
    __shared__ float2 psi[DIM];

    const int tid  = threadIdx.x;        // 0..127
    const int lane = tid & 31;           // lane within wave32
    const int wave = tid >> 5;           // 0..3
    const int b    = blockIdx.x;         // sample id
    const int r    = lane & 3;           // row of M4 this lane's A slice holds
    const int hi   = lane >> 4;          // 0: K cols 0/1 (s0), 1: K cols 2/3 (s1)
    const int pl   = lane & 15;          // pair-within-tile / B column

    // ---- init |0...0> ----
    for (int i = tid; i < DIM; i += 128) psi[i] = make_float2(0.f, 0.f);
    if (tid == 0) psi[0] = make_float2(1.f, 0.f);

    // preload per-sample embedding angles
    float xang[NQ];
    #pragma unroll
    for (int q = 0; q < NQ; ++q) xang[q] = x[b * NQ + q];

    __syncthreads();

    for (int l = 0; l < NL; ++l) {
        // ---------- single-qubit gates (WMMA f32 16x16x4) ----------
        for (int q = 0; q < NQ; ++q) {
            const float w0 = w[(l*NQ + q)*3 + 0];
            const float w1 = w[(l*NQ + q)*3 + 1];
            const float w2 = w[(l*NQ + q)*3 + 2];

            float s0, c0, s1, c1, s2, c2, sx, cx;
            __sincosf(0.5f*w0, &s0, &c0);
            __sincosf(0.5f*w1, &s1, &c1);
            __sincosf(0.5f*w2, &s2, &c2);
            __sincosf(0.5f*xang[q], &sx, &cx);

            // RX
            cpx rx00 = { c0, 0.f }, rx01 = { 0.f, -s0 };           // rx10=rx01, rx11=rx00
            // m = RY(w1) @ RX(w0)
            cpx m00 = cadd(cscl(rx00,  c1), cscl(rx01, -s1));
            cpx m01 = cadd(cscl(rx01,  c1), cscl(rx00, -s1));
            cpx m10 = cadd(cscl(rx00,  s1), cscl(rx01,  c1));
            cpx m11 = cadd(cscl(rx01,  s1), cscl(rx00,  c1));
            // f = RZ(w2) @ m   (row scaling by e^{-iw2/2}, e^{+iw2/2})
            cpx e0 = { c2, -s2 }, e1 = { c2, s2 };
            cpx f00 = cmul(e0, m00), f01 = cmul(e0, m01);
            cpx f10 = cmul(e1, m10), f11 = cmul(e1, m11);
            // G = f @ RY(x)  (RY(x) is real)
            cpx g00 = cadd(cscl(f00, cx), cscl(f01,  sx));
            cpx g01 = cadd(cscl(f01, cx), cscl(f00, -sx));
            cpx g10 = cadd(cscl(f10, cx), cscl(f11,  sx));
            cpx g11 = cadd(cscl(f11, cx), cscl(f10, -sx));

            // Real 4x4 form M4 (columns):
            //  col0=( a, b, e, f) col1=(-b, a,-f, e) col2=( c, d, g, h) col3=(-d, c,-h, g)
            // A is 16x4 = M4 stacked 4x. A VGPR0: K=(hi?2:0), VGPR1: K=(hi?3:1), row = lane&3.
            const cpx u = hi ? g01 : g00;
            const cpx v = hi ? g11 : g10;
            v2f A;
            A[0] = (r==0) ?  u.r : (r==1) ? u.i : (r==2) ?  v.r : v.i;
            A[1] = (r==0) ? -u.i : (r==1) ? u.r : (r==2) ? -v.i : v.r;

            const int bp      = 11 - q;        // wire 0 is the MSB
            const int stride  = 1 << bp;
            const int lm      = stride - 1;
            const int off     = hi * stride;   // lanes 0-15: s0, lanes 16-31: s1
            const int pb      = wave * (NTILE * 16) + pl;   // this wave's first pair

            // depth-4 software-pipelined tile loop (ring of 4 prefetch buffers)
            int    aa[PFD];
            float2 vv[PFD];
            #pragma unroll
            for (int t = 0; t < PFD; ++t) {
                aa[t] = pair_addr(pb + t * 16, bp, lm, off);
                vv[t] = psi[aa[t]];
            }

            #pragma unroll 4
            for (int t = 0; t < NTILE; ++t) {
                const int slot = t & (PFD - 1);
                const int    ac = aa[slot];
                const float2 vc = vv[slot];

                // issue next prefetch before compute (dummy self-reload on tail)
                const int tn = (t + PFD < NTILE) ? (t + PFD) : t;   // uniform select
                const int an = pair_addr(pb + tn * 16, bp, lm, off);
                vv[slot] = psi[an];
                aa[slot] = an;

                v2f B; B[0] = vc.x; B[1] = vc.y;                    // rows K0|K2=Re, K1|K3=Im
                v8f C = {0.f,0.f,0.f,0.f,0.f,0.f,0.f,0.f};
                v8f D = __builtin_amdgcn_wmma_f32_16x16x4_f32(
                            false, A, false, B, (short)0, C, false, false);

                // lanes 0-15: rows M=0/1 (Re/Im out0); lanes 16-31: rows M=10/11 (Re/Im out1)
                psi[ac] = make_float2(hi ? D[2] : D[0], hi ? D[3] : D[1]);
            }
            __syncthreads();
        }

        // ---------- CNOT ring: (0,1)...(10,11),(11,0) ----------
        for (int i = 0; i < NQ; ++i) {
            const int c  = i, t = (i + 1) % NQ;
            const int bc = 11 - c, bt = 11 - t;
            const int lo  = (bc < bt) ? bc : bt;
            const int hib = (bc < bt) ? bt : bc;
            for (int k = tid; k < 1024; k += 128) {
                int t1  = ((k  >> lo ) << (lo  + 1)) | (k  & ((1 << lo ) - 1));
                int idx = ((t1 >> hib) << (hib + 1)) | (t1 & ((1 << hib) - 1));
                idx |= (1 << bc);                 // control = 1, target bit currently 0
                const int jdx = idx | (1 << bt);
                const float2 u0 = psi[idx];
                const float2 u1 = psi[jdx];
                psi[idx] = u1;
                psi[jdx] = u0;
            }
            __syncthreads();
        }
    }

    // ---------- expectation values <X_q>,<Y_q>,<Z_q>; 3 qubits per wave ----------
    #pragma unroll
    for (int j = 0; j < 3; ++j) {
        const int q  = wave * 3 + j;
        const int bp = 11 - q;
        const int stride = 1 << bp;
        const int lm = stride - 1;
        float cr = 0.f, ci = 0.f, zz = 0.f;
        for (int p = lane; p < NPAIR; p += 32) {
            const int idx = (((p >> bp) << (bp + 1)) | (p & lm));
            const float2 a0 = psi[idx];
            const float2 a1 = psi[idx + stride];
            cr += a0.x*a1.x + a0.y*a1.y;                       // Re(conj(p0)*p1)
            ci += a0.x*a1.y - a0.y*a1.x;                       // Im(conj(p0)*p1)
            zz += a0.x*a0.x + a0.y*a0.y - a1.x*a1.x - a1.y*a1.y;
        }
        #pragma unroll
        for (int m = 16; m >= 1; m >>= 1) {
            cr += __shfl_xor(cr, m, 32);
            ci += __shfl_xor(ci, m, 32);
            zz += __shfl_xor(zz, m, 32);
        }
        if (lane == 0) {
            out[b*36 +      q] = 2.f * cr;   // <X>
            out[b*36 + 12 + q] = 2.f * ci;   // <Y>
            out[b*36 + 24 + q] = zz;         // <Z>
        }
    }
}

extern "C" void kernel_launch(void* const* d_in, const int* in_sizes, int n_in,
                              void* d_out, int out_size, void* d_ws, size_t ws_size,
                              hipStream_t stream) {
    const float* x = (const float*)d_in[0];        // [B, 12] angles
    const float* w = (const float*)d_in[1];        // [4, 12, 3] weights
    float* out = (float*)d_out;                    // [B, 36]
    const int B = in_sizes[0] / NQ;                // 8192
    qreservoir_kernel<<<B, 128, 0, stream>>>(x, w, out);
}